// FreeNet_24790551232820
// MI455X (gfx1250) — compile-verified
//
#include <hip/hip_runtime.h>

// Problem constants (match reference).
#define BATCH 1024
#define DIN   512
#define HID   512
#define DOUT  512

typedef __attribute__((ext_vector_type(2))) float v2f;
typedef __attribute__((ext_vector_type(8))) float v8f;

// ---------------------------------------------------------------------------
// FP32 WMMA GEMM:  out[M,N] = f(A[M,K] @ W[K,N] + bias[N])
//   f = relu(.)^2 when ACT, identity otherwise.
// One wave computes a 16x64 output strip: A-fragment (16x4) is reused across
// four V_WMMA_F32_16X16X4_F32 accumulators. Block = 256 threads = 8 waves.
//
// Fragment layouts (CDNA5 ISA):
//   A 16x4 (v2f): lanes 0-15 -> M=lane, K={k,k+1}; lanes 16-31 -> K={k+2,k+3}
//   B 4x16 (v2f): symmetric: lane%16 -> N, lane/16 selects K-pair
//   C/D 16x16 (v8f): lane%16 -> N, vgpr v -> M=v (+8 for lanes 16-31)
// ---------------------------------------------------------------------------
template <bool ACT>
__global__ __launch_bounds__(256) void gemm_bias_act(
    const float* __restrict__ A, const float* __restrict__ W,
    const float* __restrict__ bias, float* __restrict__ out,
    int M, int K, int N) {
  const int lane    = threadIdx.x & 31;
  const int wave    = threadIdx.x >> 5;
  const int tilesN  = N >> 6;                       // 64-wide N strips
  const int waveId  = blockIdx.x * 8 + wave;
  const int tileM   = waveId / tilesN;              // wave-uniform
  const int tileN   = waveId % tilesN;
  if (tileM * 16 >= M) return;                      // uniform per wave (EXEC stays all-1)

  const int halfSel = lane >> 4;                    // 0: lanes 0-15, 1: lanes 16-31
  const int lm      = lane & 15;
  const int koff    = halfSel * 2;
  const float* Arow = A + (size_t)(tileM * 16 + lm) * K;

  v8f acc[4] = {v8f{}, v8f{}, v8f{}, v8f{}};

  for (int k = 0; k < K; k += 4) {
    v2f a;
    a.x = Arow[k + koff];
    a.y = Arow[k + koff + 1];
#pragma unroll
    for (int t = 0; t < 4; ++t) {
      const int n = tileN * 64 + t * 16 + lm;
      v2f b;
      b.x = W[(size_t)(k + koff) * N + n];
      b.y = W[(size_t)(k + koff + 1) * N + n];
      // 8 args: (neg_a, A, neg_b, B, c_mod, C, reuse_a, reuse_b)
      acc[t] = __builtin_amdgcn_wmma_f32_16x16x4_f32(
          false, a, false, b, (short)0, acc[t], false, false);
    }
  }

#pragma unroll
  for (int t = 0; t < 4; ++t) {
    const int n  = tileN * 64 + t * 16 + lm;
    const float bv = bias[n];
#pragma unroll
    for (int v = 0; v < 8; ++v) {
      const int row = tileM * 16 + v + halfSel * 8;
      float c = acc[t][v] + bv;
      if (ACT) { c = fmaxf(c, 0.0f); c = c * c; }
      out[(size_t)row * N + n] = c;
    }
  }
}

// ---------------------------------------------------------------------------
// Blocked triangular scan.
//   h[:,j] += relu(h[:,i]*W_hh[i,j] + b_hh[i,j])^2  for j > i, i = 0..HID-2,
// with h[:,i] the FINAL value of column i.
//
// 512 threads (thread j owns column j, 16 batch rows in registers).
// Wave w owns columns [32w, 32w+32). Outer loop over 16 column blocks:
//   1) TRIANGLE (wave b only): resolve intra-block dependencies with 31
//      sequential __shfl broadcasts — pure wave-local, no barriers.
//   2) PUBLISH: wave b writes its 32x16 finalized values to ping-ponged LDS.
//   3) one __syncthreads (16 total instead of 511).
//   4) APPLY (all j >= 32(b+1)): batch-apply the 32 contributions using
//      coalesced W_hh/b_hh row loads (L2-resident) + LDS-broadcast reads.
// Buffer-reuse safety: readers of pub[b&1] (apply b) are separated from the
// next writer of pub[b&1] (publish b+2) by the barrier inside iteration b+1.
// Wave b+1 finishes its own apply(b) in program order before triangle(b+1).
// ---------------------------------------------------------------------------
#define ROWS 16
__global__ __launch_bounds__(HID) void tri_scan_blocked(
    float* __restrict__ h, const float* __restrict__ Whh,
    const float* __restrict__ bhh) {
  const int j       = threadIdx.x;          // column 0..511
  const int lane    = j & 31;
  const int wav     = j >> 5;               // wave id = column block id
  const int rowBase = blockIdx.x * ROWS;    // 16 batch rows per workgroup

  __shared__ float pub[2][32][ROWS];        // [ping-pong][i2][row]

  float hreg[ROWS];
#pragma unroll
  for (int r = 0; r < ROWS; ++r) hreg[r] = h[(size_t)(rowBase + r) * HID + j];

  for (int b = 0; b < HID / 32; ++b) {
    if (wav == b) {
      // --- TRIANGLE: finalize this wave's 32 columns (wave-local) ---
      const int colBase = b * 32;
      for (int i2 = 0; i2 < 31; ++i2) {     // i = colBase+i2 <= 510 always
        // Broadcast finalized column (colBase+i2) from lane i2; all lanes
        // of this wave execute the shuffle (branch is wave-uniform).
        float s[ROWS];
#pragma unroll
        for (int r = 0; r < ROWS; ++r) s[r] = __shfl(hreg[r], i2, 32);
        if (lane > i2) {
          const size_t off = (size_t)(colBase + i2) * HID + (colBase + lane);
          const float w  = Whh[off];
          const float bb = bhh[off];
#pragma unroll
          for (int r = 0; r < ROWS; ++r) {
            float t = fmaf(s[r], w, bb);
            t = fmaxf(t, 0.0f);
            hreg[r] = fmaf(t, t, hreg[r]);
          }
        }
      }
      // --- PUBLISH finalized block ---
#pragma unroll
      for (int r = 0; r < ROWS; ++r) pub[b & 1][lane][r] = hreg[r];
    }

    __syncthreads();

    // --- APPLY: columns beyond this block absorb its 32 contributions ---
    if (j >= (b + 1) * 32) {                // empty for b == 15
      const int iBase = b * 32;
#pragma unroll 4
      for (int i2 = 0; i2 < 32; ++i2) {     // i = iBase+i2 <= 510 when j<512
        const size_t off = (size_t)(iBase + i2) * HID + j;
        const float w  = Whh[off];
        const float bb = bhh[off];
#pragma unroll
        for (int r = 0; r < ROWS; ++r) {
          float t = fmaf(pub[b & 1][i2][r], w, bb);
          t = fmaxf(t, 0.0f);
          hreg[r] = fmaf(t, t, hreg[r]);
        }
      }
    }
  }

#pragma unroll
  for (int r = 0; r < ROWS; ++r) h[(size_t)(rowBase + r) * HID + j] = hreg[r];
}

// ---------------------------------------------------------------------------
// Launch: GEMM1 (x@W_in, relu^2) -> blocked scan -> GEMM2 (h@W_out).
// h [1024 x 512] f32 (2 MB) lives in d_ws. All launches stream-ordered.
// ---------------------------------------------------------------------------
extern "C" void kernel_launch(void* const* d_in, const int* in_sizes, int n_in,
                              void* d_out, int out_size, void* d_ws, size_t ws_size,
                              hipStream_t stream) {
  const float* x     = (const float*)d_in[0];   // [B, DIN]
  const float* W_in  = (const float*)d_in[1];   // [DIN, H]
  const float* b_in  = (const float*)d_in[2];   // [H]
  const float* W_hh  = (const float*)d_in[3];   // [H-1, H]
  const float* b_hh  = (const float*)d_in[4];   // [H-1, H]
  const float* W_out = (const float*)d_in[5];   // [H, DOUT]
  const float* b_out = (const float*)d_in[6];   // [DOUT]
  float*       out   = (float*)d_out;           // [B, DOUT]
  float*       h     = (float*)d_ws;            // [B, H] scratch

  // GEMM1: waves = (B/16)*(H/64) = 512 -> 64 blocks of 8 waves.
  gemm_bias_act<true><<<(BATCH / 16) * (HID / 64) / 8, 256, 0, stream>>>(
      x, W_in, b_in, h, BATCH, DIN, HID);

  // Blocked triangular scan: 64 blocks x 512 threads, 16 batch rows each.
  tri_scan_blocked<<<BATCH / ROWS, HID, 0, stream>>>(h, W_hh, b_hh);

  // GEMM2: out = h @ W_out + b_out.
  gemm_bias_act<false><<<(BATCH / 16) * (DOUT / 64) / 8, 256, 0, stream>>>(
      h, W_out, b_out, out, BATCH, HID, DOUT);
}